// DocQAAttention_51462298141109
// MI455X (gfx1250) — compile-verified
//
#include <hip/hip_runtime.h>

typedef float v2f __attribute__((ext_vector_type(2)));
typedef float v4f __attribute__((ext_vector_type(4)));
typedef float v8f __attribute__((ext_vector_type(8)));

#define NEGV  (-10000000.0f)
#define B_    16
#define XL_   1024
#define KL_   512
#define D_    1024
#define LDP   516   /* padded LDS row stride (floats): 516 % 64 = 4 -> conflict-free b64 frag loads */

/* workspace layout (floats) */
#define XLOG_OFF 0
#define KLOG_OFF (B_*XL_)                 /* 16384 */
#define MAXS_OFF (KLOG_OFF + B_*KL_)      /* 24576 */
#define P_OFF    (MAXS_OFF + B_*XL_)      /* 40960 */
#define K2X_OFF  (P_OFF + B_*XL_)         /* 57344 */

/* ---------------- kernel 0: row logits (x.w_input, key.w_key) ---------------- */
__global__ __launch_bounds__(256) void logits_kernel(
    const float* __restrict__ x, const float* __restrict__ key,
    const float* __restrict__ w_input, const float* __restrict__ w_key,
    float* ws) {
  const int lane = threadIdx.x & 31;
  const int gid  = blockIdx.x * 8 + (threadIdx.x >> 5);
  const float* src; const float* w; float* dst;
  if (gid < B_*XL_) { src = x + (size_t)gid * D_;              w = w_input; dst = ws + XLOG_OFF + gid; }
  else              { int g2 = gid - B_*XL_;
                      src = key + (size_t)g2 * D_;             w = w_key;   dst = ws + KLOG_OFF + g2; }
  float s = 0.f;
  #pragma unroll
  for (int j = 0; j < D_/32; ++j) { int c = lane + 32*j; s += src[c] * w[c]; }
  #pragma unroll
  for (int off = 16; off > 0; off >>= 1) s += __shfl_xor(s, off, 32);
  if (lane == 0) *dst = s;
}

/* ---------------- kernel 1: fused S -> softmax -> x2key (fp32 WMMA) ---------------- */
__global__ __launch_bounds__(256) void attn_kernel(
    const float* __restrict__ x, const float* __restrict__ x_mask,
    const float* __restrict__ key, const float* __restrict__ key_mask,
    const float* __restrict__ dot_w, float* ws, float* __restrict__ out) {
  __shared__ float Sld[16 * LDP];
  __shared__ float kmld[KL_];
  __shared__ float klld[KL_];
  __shared__ float xmld[16];
  __shared__ float xlld[16];

  const int tid  = threadIdx.x;
  const int lane = tid & 31;
  const int wv   = tid >> 5;            /* wave 0..7 */
  const int ln15 = lane & 15;
  const int koff = (lane >> 4) << 1;    /* 0 or 2 */
  const int hi8  = (lane >> 4) << 3;    /* 0 or 8 */
  const int b    = blockIdx.y;
  const int bx0  = blockIdx.x * 16;

  /* stage per-block constants */
  klld[tid]       = ws[KLOG_OFF + b*KL_ + tid];
  klld[tid + 256] = ws[KLOG_OFF + b*KL_ + tid + 256];
  kmld[tid]       = key_mask[b*KL_ + tid];
  kmld[tid + 256] = key_mask[b*KL_ + tid + 256];
  if (tid < 16) {
    xmld[tid] = x_mask[b*XL_ + bx0 + tid];
    xlld[tid] = ws[XLOG_OFF + b*XL_ + bx0 + tid];
  }
  __syncthreads();

  const float* xrow  = x   + (size_t)(b*XL_ + bx0 + ln15) * D_;
  const float* kbase = key + (size_t)b * KL_ * D_;

  /* ---- GEMM1: S(16 x 512) = (x . dot_w) @ key^T ; wave w owns cols [64w, 64w+64) ---- */
  const int n0w = wv * 64;
  v8f acc0[4] = {};
  for (int k0 = 0; k0 < D_; k0 += 4) {
    v2f xa = *(const v2f*)(xrow  + k0 + koff);
    v2f dw = *(const v2f*)(dot_w + k0 + koff);
    v2f a  = xa * dw;
    #pragma unroll
    for (int t = 0; t < 4; ++t) {
      const float* kp = kbase + (size_t)(n0w + t*16 + ln15) * D_ + k0 + koff;
      v2f bf = *(const v2f*)kp;
      acc0[t] = __builtin_amdgcn_wmma_f32_16x16x4_f32(
          false, a, false, bf, (short)0, acc0[t], false, false);
    }
  }

  /* add logits, spill raw S to LDS */
  #pragma unroll
  for (int t = 0; t < 4; ++t) {
    int n = n0w + t*16 + ln15;
    float kl = klld[n];
    #pragma unroll
    for (int i = 0; i < 8; ++i) {
      int r = i + hi8;
      Sld[r*LDP + n] = acc0[t][i] + xlld[r] + kl;
    }
  }
  __syncthreads();

  /* ---- softmax over k (2 rows per wave) + max_s ---- */
  #pragma unroll
  for (int rr = 0; rr < 2; ++rr) {
    int r = wv*2 + rr;
    float xm = xmld[r];
    float vals[16];
    float m1 = -3.4e38f, m2 = -3.4e38f;
    #pragma unroll
    for (int j = 0; j < 16; ++j) {
      int c = lane + 32*j;
      float s  = Sld[r*LDP + c];
      float km = kmld[c];
      float v1 = s + (1.f - xm*km) * NEGV;       /* masked_S */
      float v2 = s*km + (1.f - km) * NEGV;       /* S2 */
      vals[j] = v1;
      m1 = fmaxf(m1, v1); m2 = fmaxf(m2, v2);
    }
    #pragma unroll
    for (int off = 16; off > 0; off >>= 1) {
      m1 = fmaxf(m1, __shfl_xor(m1, off, 32));
      m2 = fmaxf(m2, __shfl_xor(m2, off, 32));
    }
    float ssum = 0.f;
    #pragma unroll
    for (int j = 0; j < 16; ++j) { vals[j] = __expf(vals[j] - m1); ssum += vals[j]; }
    #pragma unroll
    for (int off = 16; off > 0; off >>= 1) ssum += __shfl_xor(ssum, off, 32);
    float inv = 1.f / ssum;
    #pragma unroll
    for (int j = 0; j < 16; ++j) Sld[r*LDP + lane + 32*j] = vals[j] * inv;
    if (lane == 0) ws[MAXS_OFF + b*XL_ + bx0 + r] = m2;
  }
  __syncthreads();

  /* ---- GEMM2: x2key(16 x 1024) = attn @ key ; wave w owns d-cols [128w, 128w+128) ---- */
  const int d0 = wv * 128;
  v8f acc2[8] = {};
  for (int k0 = 0; k0 < KL_; k0 += 4) {
    v2f a = *(const v2f*)(&Sld[ln15*LDP + k0 + koff]);
    #pragma unroll
    for (int t = 0; t < 8; ++t) {
      const float* kp = kbase + (size_t)(k0 + koff) * D_ + d0 + t*16 + ln15;
      v2f bf = { kp[0], kp[D_] };
      acc2[t] = __builtin_amdgcn_wmma_f32_16x16x4_f32(
          false, a, false, bf, (short)0, acc2[t], false, false);
    }
  }

  /* epilogue: out[.,1024:2048]=x2key, out[.,2048:3072]=x*x2key */
  #pragma unroll
  for (int t = 0; t < 8; ++t) {
    int d = d0 + t*16 + ln15;
    #pragma unroll
    for (int i = 0; i < 8; ++i) {
      int r = i + hi8;
      size_t grow = (size_t)(b*XL_ + bx0 + r);
      float xv = x[grow*D_ + d];
      float v  = acc2[t][i];
      out[grow*4096 + 1024 + d] = v;
      out[grow*4096 + 2048 + d] = xv * v;
    }
  }
  /* out[.,0:1024] = x passthrough (float4 streams) */
  for (int idx = tid; idx < 16*(D_/4); idx += 256) {
    int r  = idx >> 8;
    int d4 = (idx & 255) << 2;
    size_t grow = (size_t)(b*XL_ + bx0 + r);
    *(v4f*)(out + grow*4096 + d4) = *(const v4f*)(x + grow*D_ + d4);
  }
}

/* ---------------- kernel 2a: p = norm(softmax(max_s * xm) * xm), zero key2x ---------------- */
__global__ __launch_bounds__(256) void pvec_kernel(const float* __restrict__ x_mask, float* ws) {
  __shared__ float red[256];
  const int b = blockIdx.x, tid = threadIdx.x;
  float v[4], xm[4];
  #pragma unroll
  for (int j = 0; j < 4; ++j) {
    int i = tid + 256*j;
    xm[j] = x_mask[b*XL_ + i];
    v[j]  = ws[MAXS_OFF + b*XL_ + i] * xm[j];
  }
  float m = fmaxf(fmaxf(v[0], v[1]), fmaxf(v[2], v[3]));
  red[tid] = m; __syncthreads();
  for (int off = 128; off > 0; off >>= 1) { if (tid < off) red[tid] = fmaxf(red[tid], red[tid+off]); __syncthreads(); }
  m = red[0]; __syncthreads();
  float e[4], s = 0.f;
  #pragma unroll
  for (int j = 0; j < 4; ++j) { e[j] = __expf(v[j] - m); s += e[j]; }
  red[tid] = s; __syncthreads();
  for (int off = 128; off > 0; off >>= 1) { if (tid < off) red[tid] += red[tid+off]; __syncthreads(); }
  s = red[0]; __syncthreads();
  float s2 = 0.f;
  #pragma unroll
  for (int j = 0; j < 4; ++j) { e[j] = (e[j] / s) * xm[j]; s2 += e[j]; }
  red[tid] = s2; __syncthreads();
  for (int off = 128; off > 0; off >>= 1) { if (tid < off) red[tid] += red[tid+off]; __syncthreads(); }
  s2 = red[0];
  float invn = 1.f / (s2 + 1e-13f);
  #pragma unroll
  for (int j = 0; j < 4; ++j) {
    ws[P_OFF   + b*XL_ + tid + 256*j] = e[j] * invn;
    ws[K2X_OFF + b*D_  + tid + 256*j] = 0.f;
  }
}

/* ---------------- kernel 2b: key2x[b,d] += sum_x p[b,x]*x[b,x,d] ---------------- */
__global__ __launch_bounds__(256) void key2x_kernel(const float* __restrict__ x, float* ws) {
  __shared__ float psh[64];
  const int b  = blockIdx.x >> 4;
  const int x0 = (blockIdx.x & 15) * 64;
  const int tid = threadIdx.x;
  if (tid < 64) psh[tid] = ws[P_OFF + b*XL_ + x0 + tid];
  __syncthreads();
  float acc[4] = {0.f, 0.f, 0.f, 0.f};
  for (int r = 0; r < 64; ++r) {
    float pv = psh[r];
    const float* xr = x + (size_t)(b*XL_ + x0 + r) * D_;
    #pragma unroll
    for (int j = 0; j < 4; ++j) acc[j] += pv * xr[tid + 256*j];
  }
  #pragma unroll
  for (int j = 0; j < 4; ++j) atomicAdd(&ws[K2X_OFF + b*D_ + tid + 256*j], acc[j]);
}

/* ---------------- kernel 3: out[.,3072:4096] = x * key2x(broadcast) ---------------- */
__global__ __launch_bounds__(256) void part3_kernel(const float* __restrict__ x,
                                                    const float* __restrict__ ws,
                                                    float* __restrict__ out) {
  size_t idx = (size_t)blockIdx.x * 256 + threadIdx.x;  /* one float4 each */
  int row = (int)(idx >> 8);     /* 0..16383 */
  int d4  = (int)(idx & 255) << 2;
  int b   = row >> 10;
  v4f xv = *(const v4f*)(x + (size_t)row*D_ + d4);
  v4f kv = *(const v4f*)(ws + K2X_OFF + b*D_ + d4);
  v4f r  = xv * kv;
  *(v4f*)(out + (size_t)row*4096 + 3072 + d4) = r;
}

extern "C" void kernel_launch(void* const* d_in, const int* in_sizes, int n_in,
                              void* d_out, int out_size, void* d_ws, size_t ws_size,
                              hipStream_t stream) {
  (void)in_sizes; (void)n_in; (void)out_size; (void)ws_size;
  const float* x        = (const float*)d_in[0];
  const float* x_mask   = (const float*)d_in[1];
  const float* key      = (const float*)d_in[2];
  const float* key_mask = (const float*)d_in[3];
  const float* w_input  = (const float*)d_in[4];
  const float* w_key    = (const float*)d_in[5];
  const float* dot_w    = (const float*)d_in[6];
  float* out = (float*)d_out;
  float* ws  = (float*)d_ws;

  logits_kernel<<<(B_*XL_ + B_*KL_)/8, 256, 0, stream>>>(x, key, w_input, w_key, ws);
  dim3 g1(XL_/16, B_);
  attn_kernel<<<g1, 256, 0, stream>>>(x, x_mask, key, key_mask, dot_w, ws, out);
  pvec_kernel<<<B_, 256, 0, stream>>>(x_mask, ws);
  key2x_kernel<<<B_*16, 256, 0, stream>>>(x, ws);
  part3_kernel<<<(B_*XL_*(D_/4))/256, 256, 0, stream>>>(x, ws, out);
}